// GT_10273561772511
// MI455X (gfx1250) — compile-verified
//
#include <hip/hip_runtime.h>
#include <math.h>

// ---------------------------------------------------------------------------
// Graph-attention forward for MI455X (gfx1250, wave32).
// Dense projections use V_WMMA_F32_16X16X4_F32; edge softmax/aggregation uses
// L2-resident gathers + hardware fp32 atomics.
// ---------------------------------------------------------------------------

typedef __attribute__((ext_vector_type(2))) float v2f;
typedef __attribute__((ext_vector_type(8))) float v8f;

__device__ __forceinline__ float gelu_exact(float x) {
    // torch nn.GELU default: 0.5*x*(1+erf(x/sqrt(2)))
    return 0.5f * x * (1.0f + erff(x * 0.70710678118654752f));
}

// WMMA fp32 16x16x4 fragment layouts (wave32):
//   A (16x4):  lanes 0-15 -> M=lane (K half 0), lanes 16-31 -> M=lane-16, K+2
//              vgpr0 = K {0|2}, vgpr1 = K {1|3}          -> v2f per lane
//   B (4x16):  lanes 0-15 -> N=lane (K half 0), lanes 16-31 -> K half 1
//   C/D (16x16): vgpr i -> row i (lanes 0-15) / row i+8 (lanes 16-31), col=lane&15

// ---------------------------------------------------------------------------
// Streaming-K GEMM: C[M,32] = act(A[M,K] @ W[K,32] + bias) (+resid), K large.
// One wave per 16-row tile (used only for the K=128 input projection).
// ---------------------------------------------------------------------------
template <int ACT>
__global__ __launch_bounds__(256) void gemm_n32_wmma(
    const float* __restrict__ A, int lda,
    const float* __restrict__ W,      // K x 32 row-major
    const float* __restrict__ bias,   // 32
    const float* __restrict__ resid,  // M x 32 or nullptr
    float* __restrict__ C,            // M x 32
    int M, int K)
{
    const int lane    = threadIdx.x & 31;
    const int wave    = (blockIdx.x * blockDim.x + threadIdx.x) >> 5;
    const int rowBase = wave * 16;
    if (rowBase >= M) return;  // uniform per wave

    const int lrow = lane & 15;
    const int half = lane >> 4;

    int arow = rowBase + lrow;
    if (arow >= M) arow = M - 1;  // clamp: EXEC stays all-ones for WMMA
    const float* Arow = A + (size_t)arow * lda;

    v8f acc0 = {};
    v8f acc1 = {};

    for (int kb = 0; kb < K; kb += 4) {
        const int kk = kb + 2 * half;
        v2f a = *(const v2f*)(Arow + kk);
        v2f b0, b1;
        b0.x = W[(size_t)kk * 32 + lrow];
        b0.y = W[(size_t)(kk + 1) * 32 + lrow];
        b1.x = W[(size_t)kk * 32 + 16 + lrow];
        b1.y = W[(size_t)(kk + 1) * 32 + 16 + lrow];
        acc0 = __builtin_amdgcn_wmma_f32_16x16x4_f32(
            false, a, false, b0, (short)0, acc0, false, false);
        acc1 = __builtin_amdgcn_wmma_f32_16x16x4_f32(
            false, a, false, b1, (short)0, acc1, false, false);
    }

    const int col = lrow;
    const float bc0 = bias[col];
    const float bc1 = bias[col + 16];
#pragma unroll
    for (int i = 0; i < 8; ++i) {
        const int row = rowBase + 8 * half + i;
        if (row < M) {
            float x0 = acc0[i] + bc0;
            float x1 = acc1[i] + bc1;
            if (ACT == 1) { x0 = gelu_exact(x0); x1 = gelu_exact(x1); }
            if (resid) {
                x0 += resid[(size_t)row * 32 + col];
                x1 += resid[(size_t)row * 32 + col + 16];
            }
            C[(size_t)row * 32 + col]      = x0;
            C[(size_t)row * 32 + col + 16] = x1;
        }
    }
}

// ---------------------------------------------------------------------------
// K=32 specialized GEMM: B fragments (full 32x32 weight) are preloaded into
// 32 VGPRs once per wave, then the wave grid-strides over 16-row tiles so the
// weight load is amortized. Per tile: 8 A-loads (v2f) + 16 WMMAs.
// ---------------------------------------------------------------------------
template <int ACT>
__global__ __launch_bounds__(256) void gemm_k32_wmma(
    const float* __restrict__ A,      // M x 32
    const float* __restrict__ W,      // 32 x 32 row-major
    const float* __restrict__ bias,   // 32
    const float* __restrict__ resid,  // M x 32 or nullptr
    float* __restrict__ C,            // M x 32
    int M)
{
    const int lane = threadIdx.x & 31;
    const int lrow = lane & 15;
    const int half = lane >> 4;

    // ---- loop-invariant weight fragments: 8 k-steps x 2 n-tiles ----
    v2f b[8][2];
#pragma unroll
    for (int s = 0; s < 8; ++s) {
        const int kk = 4 * s + 2 * half;
        b[s][0].x = W[kk * 32 + lrow];
        b[s][0].y = W[(kk + 1) * 32 + lrow];
        b[s][1].x = W[kk * 32 + 16 + lrow];
        b[s][1].y = W[(kk + 1) * 32 + 16 + lrow];
    }
    const float bc0 = bias[lrow];
    const float bc1 = bias[lrow + 16];

    const int waveId = (blockIdx.x * blockDim.x + threadIdx.x) >> 5;
    const int nWaves = (gridDim.x * blockDim.x) >> 5;
    const int nTiles = (M + 15) >> 4;

    for (int t = waveId; t < nTiles; t += nWaves) {
        const int rowBase = t * 16;
        int arow = rowBase + lrow;
        if (arow >= M) arow = M - 1;  // clamp: EXEC stays all-ones for WMMA
        const float* Arow = A + (size_t)arow * 32;

        v8f acc0 = {};
        v8f acc1 = {};
#pragma unroll
        for (int s = 0; s < 8; ++s) {
            v2f a = *(const v2f*)(Arow + 4 * s + 2 * half);
            acc0 = __builtin_amdgcn_wmma_f32_16x16x4_f32(
                false, a, false, b[s][0], (short)0, acc0, false, false);
            acc1 = __builtin_amdgcn_wmma_f32_16x16x4_f32(
                false, a, false, b[s][1], (short)0, acc1, false, false);
        }

        const int col = lrow;
#pragma unroll
        for (int i = 0; i < 8; ++i) {
            const int row = rowBase + 8 * half + i;
            if (row < M) {
                float x0 = acc0[i] + bc0;
                float x1 = acc1[i] + bc1;
                if (ACT == 1) { x0 = gelu_exact(x0); x1 = gelu_exact(x1); }
                if (resid) {
                    x0 += resid[(size_t)row * 32 + col];
                    x1 += resid[(size_t)row * 32 + col + 16];
                }
                C[(size_t)row * 32 + col]      = x0;
                C[(size_t)row * 32 + col + 16] = x1;
            }
        }
    }
}

// ---- monotone float<->uint encoding for atomic segment-max -----------------
__device__ __forceinline__ unsigned int float_to_ord(float f) {
    unsigned int u = __float_as_uint(f);
    return (u & 0x80000000u) ? ~u : (u | 0x80000000u);
}
__device__ __forceinline__ float ord_to_float(unsigned int u) {
    return (u & 0x80000000u) ? __uint_as_float(u ^ 0x80000000u)
                             : __uint_as_float(~u);
}

// scores[e,h] = (q[src_e,h,:] . k[dst_e,h,:]) / sqrt(8); atomicMax into smax
__global__ __launch_bounds__(256) void edge_scores_kernel(
    const float* __restrict__ q, const float* __restrict__ k,
    const int* __restrict__ src, const int* __restrict__ dst,
    float* __restrict__ scores, unsigned int* __restrict__ smax_bits, int E)
{
    const int idx = blockIdx.x * blockDim.x + threadIdx.x;
    if (idx >= E * 4) return;
    const int e = idx >> 2;
    const int h = idx & 3;
    const int s = src[e];
    const int d = dst[e];
    const float4* qp = (const float4*)(q + (size_t)s * 32 + h * 8);
    const float4* kp = (const float4*)(k + (size_t)d * 32 + h * 8);
    const float4 q0 = qp[0], q1 = qp[1];
    const float4 k0 = kp[0], k1 = kp[1];
    float sc = q0.x * k0.x + q0.y * k0.y + q0.z * k0.z + q0.w * k0.w
             + q1.x * k1.x + q1.y * k1.y + q1.z * k1.z + q1.w * k1.w;
    sc *= 0.35355339059327373f;  // 1/sqrt(HEAD_DIM=8)
    scores[idx] = sc;
    atomicMax(&smax_bits[(size_t)d * 4 + h], float_to_ord(sc));
}

// scores[e,h] = exp(scores - smax[dst]); atomicAdd into denom[dst]
__global__ __launch_bounds__(256) void edge_expsum_kernel(
    float* __restrict__ scores, const unsigned int* __restrict__ smax_bits,
    float* __restrict__ denom, const int* __restrict__ dst, int E)
{
    const int idx = blockIdx.x * blockDim.x + threadIdx.x;
    if (idx >= E * 4) return;
    const int e = idx >> 2;
    const int h = idx & 3;
    const int d = dst[e];
    const float m  = ord_to_float(smax_bits[(size_t)d * 4 + h]);
    const float ev = expf(scores[idx] - m);
    scores[idx] = ev;
    atomicAdd(&denom[(size_t)d * 4 + h], ev);
}

// attn[dst] += v[src] * (scores / denom[dst])
__global__ __launch_bounds__(256) void edge_scatter_kernel(
    const float* __restrict__ scores, const float* __restrict__ denom,
    const float* __restrict__ v, const int* __restrict__ src,
    const int* __restrict__ dst, float* __restrict__ attn, int E)
{
    const int idx = blockIdx.x * blockDim.x + threadIdx.x;
    if (idx >= E * 4) return;
    const int e = idx >> 2;
    const int h = idx & 3;
    const int s = src[e];
    const int d = dst[e];
    const float p = scores[idx] / denom[(size_t)d * 4 + h];
    const float4* vp = (const float4*)(v + (size_t)s * 32 + h * 8);
    const float4 v0 = vp[0], v1 = vp[1];
    float* ap = attn + (size_t)d * 32 + h * 8;
    atomicAdd(ap + 0, v0.x * p);
    atomicAdd(ap + 1, v0.y * p);
    atomicAdd(ap + 2, v0.z * p);
    atomicAdd(ap + 3, v0.w * p);
    atomicAdd(ap + 4, v1.x * p);
    atomicAdd(ap + 5, v1.y * p);
    atomicAdd(ap + 6, v1.z * p);
    atomicAdd(ap + 7, v1.w * p);
}

// out[M,2] = h[M,32] @ W_out[32,2] + b_out
__global__ __launch_bounds__(256) void classify_kernel(
    const float* __restrict__ h, const float* __restrict__ Wout,
    const float* __restrict__ bout, float* __restrict__ out, int M)
{
    const int n = blockIdx.x * blockDim.x + threadIdx.x;
    if (n >= M) return;
    float a0 = bout[0];
    float a1 = bout[1];
    const float* hr = h + (size_t)n * 32;
#pragma unroll
    for (int j = 0; j < 32; ++j) {
        const float hv = hr[j];
        a0 += hv * Wout[j * 2 + 0];
        a1 += hv * Wout[j * 2 + 1];
    }
    out[(size_t)n * 2 + 0] = a0;
    out[(size_t)n * 2 + 1] = a1;
}

extern "C" void kernel_launch(void* const* d_in, const int* in_sizes, int n_in,
                              void* d_out, int out_size, void* d_ws, size_t ws_size,
                              hipStream_t stream)
{
    const float* x     = (const float*)d_in[0];
    const int*   src   = (const int*)  d_in[1];
    const int*   dst   = (const int*)  d_in[2];
    const float* W_in  = (const float*)d_in[3];
    const float* b_in  = (const float*)d_in[4];
    const float* Wq    = (const float*)d_in[5];
    const float* bq    = (const float*)d_in[6];
    const float* Wk    = (const float*)d_in[7];
    const float* bk    = (const float*)d_in[8];
    const float* Wv    = (const float*)d_in[9];
    const float* bv    = (const float*)d_in[10];
    const float* Wo    = (const float*)d_in[11];
    const float* bo    = (const float*)d_in[12];
    const float* W1    = (const float*)d_in[13];
    const float* b1    = (const float*)d_in[14];
    const float* W2    = (const float*)d_in[15];
    const float* b2    = (const float*)d_in[16];
    const float* Wout  = (const float*)d_in[17];
    const float* bout  = (const float*)d_in[18];
    float* out = (float*)d_out;

    const int N = in_sizes[0] / 128;   // 100000
    const int E = in_sizes[1];         // 3200000

    // ---- workspace carve (256B aligned) ----
    char* w = (char*)d_ws;
    auto carve = [&](size_t bytes) -> void* {
        void* p = (void*)w;
        w += (bytes + 255) & ~(size_t)255;
        return p;
    };
    const size_t nodeF = (size_t)N * 32 * sizeof(float);
    float*        hbuf   = (float*)       carve(nodeF);
    float*        qbuf   = (float*)       carve(nodeF);
    float*        kbuf   = (float*)       carve(nodeF);
    float*        vbuf   = (float*)       carve(nodeF);
    float*        attn   = (float*)       carve(nodeF);           // reused as FFN hidden
    float*        scores = (float*)       carve((size_t)E * 4 * sizeof(float));
    unsigned int* smax   = (unsigned int*)carve((size_t)N * 4 * sizeof(unsigned int));
    float*        denom  = (float*)       carve((size_t)N * 4 * sizeof(float));
    (void)ws_size;

    const int TB = 256;
    const int gemmBlocks = (((N + 15) / 16) * 32 + TB - 1) / TB;  // 1 wave / 16 rows
    const int k32Blocks  = 256;   // 2048 waves grid-striding ~6250 tiles (~3 tiles/wave)
    const int edgeBlocks = (E * 4 + TB - 1) / TB;
    const int nodeBlocks = (N + TB - 1) / TB;

    // h = gelu(x @ W_in + b_in)   (K=128 streaming version)
    gemm_n32_wmma<1><<<gemmBlocks, TB, 0, stream>>>(x, 128, W_in, b_in, nullptr,
                                                    hbuf, N, 128);

    for (int l = 0; l < 2; ++l) {
        const float* Wq_l = Wq + l * 32 * 32; const float* bq_l = bq + l * 32;
        const float* Wk_l = Wk + l * 32 * 32; const float* bk_l = bk + l * 32;
        const float* Wv_l = Wv + l * 32 * 32; const float* bv_l = bv + l * 32;
        const float* Wo_l = Wo + l * 32 * 32; const float* bo_l = bo + l * 32;
        const float* W1_l = W1 + l * 32 * 32; const float* b1_l = b1 + l * 32;
        const float* W2_l = W2 + l * 32 * 32; const float* b2_l = b2 + l * 32;

        // q, k, v projections (K=32, register-cached weights)
        gemm_k32_wmma<0><<<k32Blocks, TB, 0, stream>>>(hbuf, Wq_l, bq_l,
                                                       nullptr, qbuf, N);
        gemm_k32_wmma<0><<<k32Blocks, TB, 0, stream>>>(hbuf, Wk_l, bk_l,
                                                       nullptr, kbuf, N);
        gemm_k32_wmma<0><<<k32Blocks, TB, 0, stream>>>(hbuf, Wv_l, bv_l,
                                                       nullptr, vbuf, N);

        // reset per-layer accumulators (graph-capturable async memsets)
        hipMemsetAsync(smax,  0, (size_t)N * 4 * sizeof(unsigned int), stream);
        hipMemsetAsync(denom, 0, (size_t)N * 4 * sizeof(float), stream);
        hipMemsetAsync(attn,  0, nodeF, stream);

        // edge softmax + aggregation
        edge_scores_kernel <<<edgeBlocks, TB, 0, stream>>>(qbuf, kbuf, src, dst,
                                                           scores, smax, E);
        edge_expsum_kernel <<<edgeBlocks, TB, 0, stream>>>(scores, smax, denom,
                                                           dst, E);
        edge_scatter_kernel<<<edgeBlocks, TB, 0, stream>>>(scores, denom, vbuf,
                                                           src, dst, attn, E);

        // h = h + attn @ Wo + bo   (element-wise resid: in-place safe)
        gemm_k32_wmma<0><<<k32Blocks, TB, 0, stream>>>(attn, Wo_l, bo_l,
                                                       hbuf, hbuf, N);
        // ffn hidden = gelu(h @ W1 + b1)   (reuse attn buffer)
        gemm_k32_wmma<1><<<k32Blocks, TB, 0, stream>>>(hbuf, W1_l, b1_l,
                                                       nullptr, attn, N);
        // h = h + hidden @ W2 + b2
        gemm_k32_wmma<0><<<k32Blocks, TB, 0, stream>>>(attn, W2_l, b2_l,
                                                       hbuf, hbuf, N);
    }

    // out = h @ W_out + b_out
    classify_kernel<<<nodeBlocks, TB, 0, stream>>>(hbuf, Wout, bout, out, N);
}